// Decoder_360777253259
// MI455X (gfx1250) — compile-verified
//
#include <hip/hip_runtime.h>
#include <hip/hip_bf16.h>

// Problem sizes (fixed by the reference)
#define H     1024
#define B     64
#define T     256
#define WPAD  (H + 4)   // LDS row stride in floats: bank-conflict-free for b64 reads

typedef __attribute__((ext_vector_type(2))) float v2f;
typedef __attribute__((ext_vector_type(4))) float v4f;
typedef __attribute__((ext_vector_type(8))) float v8f;

// D = A(16x4 f32) * B(4x16 f32) + C(16x16 f32), full fp32 precision (matches reference math)
__device__ __forceinline__ v8f wmma4(v2f a, v2f b, v8f c) {
    return __builtin_amdgcn_wmma_f32_16x16x4_f32(
        /*neg_a=*/false, a, /*neg_b=*/false, b,
        /*c_mod=*/(short)0, c, /*reuse_a=*/false, /*reuse_b=*/false);
}

__device__ __forceinline__ float sigf(float x) { return 1.0f / (1.0f + __expf(-x)); }

// Persistent LSTM scan kernel.
// Grid: 64 blocks x 256 threads (8 wave32). Block ct owns hidden columns
// [16*ct, 16*ct+16) and the 4 matching gate-column tiles of W_ih / W_hh.
// CDNA5-specific: the step-invariant W_hh slice (256 KB) lives in the 320 KB
// LDS for the whole scan, so each timestep only streams W_ih from L2 while
// the recurrent operand comes from ds_load_b64. Cell state c also lives in
// LDS; h is exchanged through L2 with a device-wide atomic barrier per step.
__global__ __launch_bounds__(256, 1)
void lstm_persistent(const int*   __restrict__ tgt,   // [B,T] token ids
                     const float* __restrict__ h0,    // [B,H]
                     const float* __restrict__ c0,    // [B,H]
                     const float* __restrict__ emb,   // [VOCAB,H]
                     const float* __restrict__ Wih,   // [4H,H]
                     const float* __restrict__ Whh,   // [4H,H]
                     const float* __restrict__ bih,   // [4H]
                     const float* __restrict__ bhh,   // [4H]
                     float*       __restrict__ out,   // [B,T,H] ++ h_fin ++ c_fin
                     float*       __restrict__ hbuf,  // [2][B*H] exchange buffers
                     unsigned*    __restrict__ counter)
{
    __shared__ float ldsW[64 * WPAD];   // W_hh slice: (4 gates x 16 cols) x H, ~257 KB
    __shared__ float ldsG[4][B * 16];   // gate tiles (i,f,g,o) for this column slice: 16 KB
    __shared__ float ldsC[B * 16];      // persistent cell state slice: 4 KB

    const int tid  = threadIdx.x;
    const int lane = tid & 31;
    const int wave = tid >> 5;
    const int n0   = blockIdx.x * 16;   // base hidden column of this block

    // ---- one-time: stage this block's W_hh slice into LDS (float4 copies)
    for (int i4 = tid; i4 < 64 * (H / 4); i4 += 256) {
        const int row = i4 >> 8;            // 0..63 = gate*16 + col
        const int k4  = (i4 & 255) * 4;     // k offset
        const int gcol = (row >> 4) * H + n0 + (row & 15);
        const v4f w = *(const v4f*)(Whh + (size_t)gcol * H + k4);
        *(v4f*)(ldsW + row * WPAD + k4) = w;
    }
    // ---- one-time: init persistent c slice from c0
    for (int e = tid; e < B * 16; e += 256)
        ldsC[e] = c0[(e >> 4) * H + n0 + (e & 15)];
    __syncthreads();

    float* const hb0  = hbuf;
    float* const hb1  = hbuf + B * H;
    float* const hfin = out + (size_t)B * T * H;
    float* const cfin = hfin + B * H;

    const int klane = 2 * (lane >> 4);  // per-lane K phase for 16x16x4 A/B operands

    for (int t = 0; t < T; ++t) {
        const float* hprev = (t == 0) ? h0 : (((t & 1) == 0) ? hb1 : hb0);
        float*       hnext = ((t & 1) == 0) ? hb0 : hb1;

        // ---- matrix phase: each wave computes 2 of the 16 (gate, M-tile) 16x16 tiles
        for (int r = 0; r < 2; ++r) {
            const int combo = wave + 8 * r;       // 0..15
            const int gate  = combo >> 2;         // 0..3  (i,f,g,o)
            const int mt    = combo & 3;          // 0..3  (batch row tile)
            const int m     = mt * 16 + (lane & 15);
            const int tok   = tgt[m * T + t];     // embedding row for batch m at time t
            const float* xrow = emb + (size_t)tok * H;
            const float* hrow = hprev + m * H;
            const float* wihr = Wih + (size_t)(gate * H + n0 + (lane & 15)) * H;
            const float* lwhh = ldsW + (gate * 16 + (lane & 15)) * WPAD;

            v8f acc = {};
            #pragma unroll 8
            for (int k0 = 0; k0 < H; k0 += 4) {
                const int kk = k0 + klane;
                v2f ax = *(const v2f*)(xrow + kk);   // A: x_t rows           (global/L2)
                v2f bx = *(const v2f*)(wihr + kk);   // B: W_ih^T columns     (global/L2)
                acc = wmma4(ax, bx, acc);
                v2f ah = *(const v2f*)(hrow + kk);   // A: h_{t-1} rows       (global/L2)
                v2f bh = *(const v2f*)(lwhh + kk);   // B: W_hh^T columns     (LDS, resident)
                acc = wmma4(ah, bh, acc);
            }
            // scatter C/D tile (M = v + 8*(lane>>4), N = lane&15) into LDS
            #pragma unroll
            for (int v = 0; v < 8; ++v) {
                const int row = mt * 16 + v + 8 * (lane >> 4);
                ldsG[gate][row * 16 + (lane & 15)] = acc[v];
            }
        }
        __syncthreads();

        // ---- elementwise LSTM cell on the 64x16 slice
        for (int e = tid; e < B * 16; e += 256) {
            const int row = e >> 4;
            const int hc  = n0 + (e & 15);
            const float iv = ldsG[0][e] + bih[0 * H + hc] + bhh[0 * H + hc];
            const float fv = ldsG[1][e] + bih[1 * H + hc] + bhh[1 * H + hc];
            const float gv = ldsG[2][e] + bih[2 * H + hc] + bhh[2 * H + hc];
            const float ov = ldsG[3][e] + bih[3 * H + hc] + bhh[3 * H + hc];
            const float cp = ldsC[e];
            const float cn = sigf(fv) * cp + sigf(iv) * tanhf(gv);
            const float hn = sigf(ov) * tanhf(cn);
            ldsC[e] = cn;
            hnext[row * H + hc] = hn;                 // re-read next step: regular (RT) store
            // outputs are write-once: non-temporal so they don't evict L2-resident weights
            __builtin_nontemporal_store(hn, &out[((size_t)row * T + t) * H + hc]);
            if (t == T - 1) {
                hfin[row * H + hc] = hn;
                cfin[row * H + hc] = cn;
            }
        }

        // ---- device-wide barrier (monotonic counter; memset to 0 per launch)
        __threadfence();
        __syncthreads();
        if (tid == 0) {
            __hip_atomic_fetch_add(counter, 1u, __ATOMIC_RELEASE, __HIP_MEMORY_SCOPE_AGENT);
            const unsigned target = (unsigned)(t + 1) * gridDim.x;
            while (__hip_atomic_load(counter, __ATOMIC_ACQUIRE, __HIP_MEMORY_SCOPE_AGENT) < target) {
                __builtin_amdgcn_s_sleep(8);
            }
        }
        __syncthreads();
        __threadfence();
    }
}

extern "C" void kernel_launch(void* const* d_in, const int* in_sizes, int n_in,
                              void* d_out, int out_size, void* d_ws, size_t ws_size,
                              hipStream_t stream) {
    const int*   tgt = (const int*)  d_in[0];   // [B,T]
    const float* h0  = (const float*)d_in[1];   // [1,B,H]
    const float* c0  = (const float*)d_in[2];   // [1,B,H]
    // d_in[3] encoder_outputs, d_in[4] src_lengths: unused in the math
    const float* emb = (const float*)d_in[5];   // [VOCAB,H]
    const float* Wih = (const float*)d_in[6];   // [4H,H]
    const float* Whh = (const float*)d_in[7];   // [4H,H]
    const float* bih = (const float*)d_in[8];   // [4H]
    const float* bhh = (const float*)d_in[9];   // [4H]

    float*    out     = (float*)d_out;
    unsigned* counter = (unsigned*)d_ws;                       // barrier counter @ ws[0]
    float*    hbuf    = (float*)((char*)d_ws + 256);           // 2 x B*H fp32 = 512 KB

    (void)in_sizes; (void)n_in; (void)out_size; (void)ws_size;

    hipMemsetAsync(d_ws, 0, 256, stream);                      // reset barrier counter
    lstm_persistent<<<64, 256, 0, stream>>>(tgt, h0, c0, emb, Wih, Whh, bih, bhh,
                                            out, hbuf, counter);
}